// Blur_88055419503066
// MI455X (gfx1250) — compile-verified
//
#include <hip/hip_runtime.h>
#include <math.h>

typedef __attribute__((ext_vector_type(2))) float v2f;
typedef __attribute__((ext_vector_type(8))) float v8f;

#define HH 128
#define WW 128
#define PLANE (HH*WW)
#define TPW 16   // tiles per wave

// One wave (32 lanes) per block. Each wave processes TPW 16x16 output tiles.
// out[m,n] = sum_dy sum_k A_dy[m,k] * B_dy[k,n]
//   A_dy[m,k] = x[ybase+m+dy-2][xbase+k-2]        (k in [0,20), chunks of 4)
//   B_dy[k,n] = w2d[dy][k-n] for 0<=k-n<=4 else 0 (Toeplitz band of weights)
__global__ __launch_bounds__(32) void blur5_wmma_f32(
    const float* __restrict__ src,
    const float* __restrict__ logls,
    float* __restrict__ dst,
    int nPlanes)
{
    const int lane = threadIdx.x & 31;
    const int m    = lane & 15;   // A row / B col / D col
    const int half = lane >> 4;
    const int kb   = half << 1;   // per-lane K offset within a K=4 chunk

    // ---- build normalized weights and all 25 Toeplitz B fragments (once) ----
    v2f bf[5][5];
    {
        const float inv2ls = 0.5f * expf(-logls[0]);   // 1/(2*exp(ll))
        float wr[5][5];
        float s = 0.0f;
        #pragma unroll
        for (int i = 0; i < 5; ++i) {
            #pragma unroll
            for (int j = 0; j < 5; ++j) {
                float di = (float)(i - 2), dj = (float)(j - 2);
                float q  = di * di + dj * dj;
                float e  = expf(-(q * q) * inv2ls);
                wr[i][j] = e;
                s += e;
            }
        }
        const float inv = 1.0f / s;
        #pragma unroll
        for (int dy = 0; dy < 5; ++dy) {
            #pragma unroll
            for (int ch = 0; ch < 5; ++ch) {
                const int tlo = ch * 4 + kb - m;   // tap index for .x ; .y is tlo+1
                float blo = 0.0f, bhi = 0.0f;
                #pragma unroll
                for (int t = 0; t < 5; ++t) {
                    const float wv = wr[dy][t] * inv;
                    blo = (tlo == t)     ? wv : blo;
                    bhi = (tlo + 1 == t) ? wv : bhi;
                }
                v2f b; b.x = blo; b.y = bhi;
                bf[dy][ch] = b;
            }
        }
    }

    const long totalTiles = (long)nPlanes * 64;   // 8x8 tiles per 128x128 plane
    const long base = (long)blockIdx.x * TPW;

    for (int i = 0; i < TPW; ++i) {
        const long tid = base + i;
        if (tid >= totalTiles) break;             // wave-uniform
        const int plane = (int)(tid >> 6);
        const int ty    = (int)((tid >> 3) & 7);
        const int tx    = (int)(tid & 7);
        const int ybase = ty << 4;
        const int xbase = tx << 4;

        const float* __restrict__ sp = src + (size_t)plane * PLANE;
        v8f acc = {};

        const bool interior = (tx >= 1) & (tx <= 6) & (ty >= 1) & (ty <= 6); // uniform
        if (interior) {
            #pragma unroll
            for (int dy = 0; dy < 5; ++dy) {
                const float* __restrict__ rp =
                    sp + (size_t)(ybase + m + dy - 2) * WW + (xbase + kb - 2);
                #pragma unroll
                for (int ch = 0; ch < 5; ++ch) {
                    const v2f a = *(const v2f*)(rp + ch * 4);   // 8B-aligned b64 load
                    acc = __builtin_amdgcn_wmma_f32_16x16x4_f32(
                        false, a, false, bf[dy][ch], (short)0, acc, false, false);
                }
            }
        } else {
            #pragma unroll
            for (int dy = 0; dy < 5; ++dy) {
                const int  r   = ybase + m + dy - 2;
                const bool rok = (r >= 0) & (r < HH);
                const int  rc  = min(max(r, 0), HH - 1);
                const float* __restrict__ rp = sp + (size_t)rc * WW;
                #pragma unroll
                for (int ch = 0; ch < 5; ++ch) {
                    const int c0 = xbase + ch * 4 + kb - 2;
                    const int c1 = c0 + 1;
                    const int c0c = min(max(c0, 0), WW - 1);
                    const int c1c = min(max(c1, 0), WW - 1);
                    const float vx = rp[c0c];
                    const float vy = rp[c1c];
                    v2f a;
                    a.x = (rok & (c0 >= 0) & (c0 < WW)) ? vx : 0.0f;
                    a.y = (rok & (c1 >= 0) & (c1 < WW)) ? vy : 0.0f;
                    acc = __builtin_amdgcn_wmma_f32_16x16x4_f32(
                        false, a, false, bf[dy][ch], (short)0, acc, false, false);
                }
            }
        }

        // ---- store D: lane l VGPR v -> (row = v + 8*half, col = m) ----
        float* __restrict__ dp =
            dst + (size_t)plane * PLANE + (size_t)ybase * WW + xbase;
        #pragma unroll
        for (int v = 0; v < 8; ++v) {
            dp[(size_t)(v + 8 * half) * WW + m] = acc[v];
        }
    }
}

extern "C" void kernel_launch(void* const* d_in, const int* in_sizes, int n_in,
                              void* d_out, int out_size, void* d_ws, size_t ws_size,
                              hipStream_t stream) {
    const float* x   = (const float*)d_in[0];
    const float* ll  = (const float*)d_in[1];
    float*       out = (float*)d_out;

    const int nPlanes = in_sizes[0] / PLANE;            // 32*128 = 4096
    const long totalTiles = (long)nPlanes * 64;         // 262144
    const int blocks = (int)((totalTiles + TPW - 1) / TPW);

    blur5_wmma_f32<<<blocks, 32, 0, stream>>>(x, ll, out, nPlanes);
}